// SelfAttention_56959856279668
// MI455X (gfx1250) — compile-verified
//
#include <hip/hip_runtime.h>
#include <hip/hip_bf16.h>
#include <math.h>

typedef __attribute__((ext_vector_type(2))) float v2f;
typedef __attribute__((ext_vector_type(8))) float v8f;

#define B_  2
#define S_  2048
#define H_  1024
#define NH_ 16
#define DH_ 64

// ---------------------------------------------------------------------------
// Kernel 0: pair-swizzle the weight matrices so the WMMA B-operand pair
// (W[k][n], W[k+1][n]) is contiguous:  Wp[(k>>1)*2*H + n*2 + (k&1)] = W[k][n].
// One block per (mat, k-row); reads coalesced, 24MB one-off traffic.
// ---------------------------------------------------------------------------
__global__ __launch_bounds__(256)
void swizzle_w_kernel(const float* __restrict__ Wq, const float* __restrict__ Wk,
                      const float* __restrict__ Wv, float* __restrict__ Wp)
{
    const int k   = blockIdx.x & (H_ - 1);
    const int mat = blockIdx.x >> 10;
    const float* W = (mat == 0) ? Wq : (mat == 1) ? Wk : Wv;
    float* dst = Wp + (size_t)mat * H_ * H_;

    const int t = threadIdx.x;
    const float4 v = *(const float4*)(W + k * H_ + t * 4);
    const int base = (k >> 1) * (2 * H_) + (k & 1);
    dst[base + (t * 4 + 0) * 2] = v.x;
    dst[base + (t * 4 + 1) * 2] = v.y;
    dst[base + (t * 4 + 2) * 2] = v.z;
    dst[base + (t * 4 + 3) * 2] = v.w;
}

// ---------------------------------------------------------------------------
// Kernel 1: fused QKV projections.  grid (256, 6), block 256 (8 waves).
// Each wave computes a 16x64 slab (4 N-tiles) of one of Q/K/V so one A-operand
// ds_load_b64 feeds four V_WMMA_F32_16X16X4_F32; B-operands are single
// coalesced b64 loads from the pair-swizzled Wp.  Q/K stored head-split
// [B,NH,S,DH]; V stored pair-swizzled per head for phase C b64 loads.
// ---------------------------------------------------------------------------
__global__ __launch_bounds__(256)
void qkv_proj_kernel(const float* __restrict__ X,  const float* __restrict__ Wp,
                     const float* __restrict__ bq, const float* __restrict__ bk,
                     const float* __restrict__ bv,
                     float* __restrict__ Qh, float* __restrict__ Kh,
                     float* __restrict__ Vp)
{
    __shared__ float xs[16 * 260];           // 16 rows x 256-K chunk (+4 pad)

    const int tid  = threadIdx.x;
    const int lane = tid & 31;
    const int wave = tid >> 5;
    const int hi   = lane >> 4;              // half-wave select (K +0 / +2)
    const int ln   = lane & 15;

    const int m0   = blockIdx.x * 16;        // output row tile (B*S rows)
    const int mat  = blockIdx.y >> 1;        // 0=Q 1=K 2=V
    const int half = blockIdx.y & 1;
    const int nwb  = half * 512 + wave * 64; // wave's 64-col slab (64-aligned)

    const float* W    = Wp + (size_t)mat * H_ * H_;
    const float* bias = (mat == 0) ? bq : (mat == 1) ? bk : bv;

    v8f cacc[4];
    #pragma unroll
    for (int n = 0; n < 4; ++n)
        #pragma unroll
        for (int j = 0; j < 8; ++j) cacc[n][j] = 0.0f;

    const int lrow = tid >> 4;               // cooperative-load row
    const int lc4  = tid & 15;               // cooperative-load column group

    for (int kc = 0; kc < H_; kc += 256) {
        __syncthreads();
        #pragma unroll
        for (int j = 0; j < 4; ++j) {
            const float4 v =
                *(const float4*)(X + (m0 + lrow) * H_ + kc + lc4 * 16 + j * 4);
            *(float4*)(xs + lrow * 260 + lc4 * 16 + j * 4) = v;
        }
        __syncthreads();

        #pragma unroll 4
        for (int k = 0; k < 256; k += 4) {
            const v2f a = *(const v2f*)(xs + ln * 260 + k + 2 * hi); // A: X[m][k]
            const int kk = kc + k + 2 * hi;                          // even
            const float* wrow = W + (kk >> 1) * (2 * H_) + 2 * ln;
            #pragma unroll
            for (int nt = 0; nt < 4; ++nt) {
                const v2f bw = *(const v2f*)(wrow + (nwb + nt * 16) * 2);
                cacc[nt] = __builtin_amdgcn_wmma_f32_16x16x4_f32(
                    false, a, false, bw, (short)0, cacc[nt], false, false);
            }
        }
    }

    const int h = nwb >> 6;                  // head index (slab is 64-aligned)
    #pragma unroll
    for (int nt = 0; nt < 4; ++nt) {
        const float bval = bias[nwb + nt * 16 + ln];
        const int d = nt * 16 + ln;
        #pragma unroll
        for (int j = 0; j < 8; ++j) {
            const int m  = m0 + j + hi * 8;  // C layout: row = j + 8*hi
            const int bi = m >> 11;          // batch
            const int s  = m & (S_ - 1);
            const float val = cacc[nt][j] + bval;
            if (mat == 0)
                Qh[((bi * NH_ + h) * S_ + s) * DH_ + d] = val;
            else if (mat == 1)
                Kh[((bi * NH_ + h) * S_ + s) * DH_ + d] = val;
            else
                Vp[(size_t)(bi * NH_ + h) * S_ * DH_ +
                   (s >> 1) * (2 * DH_) + d * 2 + (s & 1)] = val;
        }
    }
}

// ---------------------------------------------------------------------------
// Kernel 2: attention.  grid (S/16, NH, B), block 256 (8 waves).
// Workgroup handles one (b, h, 16-query-row) slab.  Each wave owns 256 key
// columns: scores kept in 16 v8f accumulators, softmax via shfl_xor + small
// LDS reductions, attn written coalesced, then P·V with a per-wave LDS
// transpose stage and single-b64 swizzled-V B-operands; ctx reduced in LDS.
// ---------------------------------------------------------------------------
__global__ __launch_bounds__(256)
void attention_kernel(const float* __restrict__ Qh, const float* __restrict__ Kh,
                      const float* __restrict__ Vp, const float* __restrict__ mask,
                      float* __restrict__ ctx, float* __restrict__ attn)
{
    __shared__ float qs[16 * 68];            // Q tile 16x64 (+4 pad)
    __shared__ float redr[128];              // per-wave row max / sum
    __shared__ float myPs[8][16 * 20];       // per-wave P-tile transpose stage
    __shared__ float red[8][16 * 68];        // per-wave ctx partials

    const int tid  = threadIdx.x;
    const int lane = tid & 31;
    const int wave = tid >> 5;
    const int hi   = lane >> 4;
    const int ln   = lane & 15;

    const int q0 = blockIdx.x * 16;
    const int h  = blockIdx.y;
    const int b  = blockIdx.z;
    const int hb = (b * NH_ + h) * S_;                   // row base (Qh/Kh)
    const size_t hbD = (size_t)(b * NH_ + h) * S_ * DH_; // elem base (Vp)

    // ---- stage Q tile in LDS ----
    {
        const int row = tid >> 4, seg = tid & 15;
        *(float4*)(qs + row * 68 + seg * 4) =
            *(const float4*)(Qh + (hb + q0 + row) * DH_ + seg * 4);
    }
    __syncthreads();

    // ---- phase A: scores for this wave's 256 columns (16 tiles) ----
    const int kwb = wave * 256;
    v8f   sc[16];
    float rmax[8];
    #pragma unroll
    for (int j = 0; j < 8; ++j) rmax[j] = -3.402823466e38f;

    #pragma unroll
    for (int t = 0; t < 16; ++t) {
        const int kb = kwb + t * 16;
        v8f acc = {};
        #pragma unroll
        for (int k = 0; k < DH_; k += 4) {
            const v2f a  = *(const v2f*)(qs + ln * 68 + k + 2 * hi);
            const v2f bb = *(const v2f*)(Kh + (hb + kb + ln) * DH_ + k + 2 * hi);
            acc = __builtin_amdgcn_wmma_f32_16x16x4_f32(false, a, false, bb,
                                                        (short)0, acc, false, false);
        }
        const float mw = mask[b * S_ + kb + ln] * 0.125f;  // mask * 1/sqrt(64)
        #pragma unroll
        for (int j = 0; j < 8; ++j) {
            const float v = acc[j] * mw;
            acc[j]  = v;
            rmax[j] = fmaxf(rmax[j], v);
        }
        sc[t] = acc;
    }

    // ---- row max: butterfly across 16-lane row groups, then across waves ----
    #pragma unroll
    for (int j = 0; j < 8; ++j) {
        #pragma unroll
        for (int m = 8; m >= 1; m >>= 1)
            rmax[j] = fmaxf(rmax[j], __shfl_xor(rmax[j], m, 32));
    }
    if (ln == 0) {
        #pragma unroll
        for (int j = 0; j < 8; ++j) redr[wave * 16 + hi * 8 + j] = rmax[j];
    }
    __syncthreads();
    float gmax[8];
    #pragma unroll
    for (int j = 0; j < 8; ++j) {
        float m = -3.402823466e38f;
        #pragma unroll
        for (int w = 0; w < 8; ++w) m = fmaxf(m, redr[w * 16 + hi * 8 + j]);
        gmax[j] = m;
    }
    __syncthreads();

    // ---- exp + row sum ----
    float rsum[8];
    #pragma unroll
    for (int j = 0; j < 8; ++j) rsum[j] = 0.0f;
    #pragma unroll
    for (int t = 0; t < 16; ++t) {
        #pragma unroll
        for (int j = 0; j < 8; ++j) {
            const float p = __expf(sc[t][j] - gmax[j]);
            sc[t][j] = p;
            rsum[j] += p;
        }
    }
    #pragma unroll
    for (int j = 0; j < 8; ++j) {
        #pragma unroll
        for (int m = 8; m >= 1; m >>= 1)
            rsum[j] += __shfl_xor(rsum[j], m, 32);
    }
    if (ln == 0) {
        #pragma unroll
        for (int j = 0; j < 8; ++j) redr[wave * 16 + hi * 8 + j] = rsum[j];
    }
    __syncthreads();
    float inv[8];
    #pragma unroll
    for (int j = 0; j < 8; ++j) {
        float ssum = 0.0f;
        #pragma unroll
        for (int w = 0; w < 8; ++w) ssum += redr[w * 16 + hi * 8 + j];
        inv[j] = 1.0f / ssum;
    }

    // ---- normalize + write attn (coalesced 64B runs per half-wave) ----
    const size_t abase = ((size_t)(b * NH_ + h) * S_ + q0) * S_;
    #pragma unroll
    for (int t = 0; t < 16; ++t) {
        const int col = kwb + t * 16 + ln;
        #pragma unroll
        for (int j = 0; j < 8; ++j) {
            const int row = j + hi * 8;
            const float p = sc[t][j] * inv[j];
            sc[t][j] = p;
            attn[abase + (size_t)row * S_ + col] = p;
        }
    }

    // ---- phase C: partial ctx = P(16x256) @ V(256x64) per wave ----
    v8f cacc[4];
    #pragma unroll
    for (int n = 0; n < 4; ++n)
        #pragma unroll
        for (int j = 0; j < 8; ++j) cacc[n][j] = 0.0f;

    float* ps = &myPs[wave][0];              // private per-wave stage (stride 20)
    #pragma unroll
    for (int t = 0; t < 16; ++t) {
        const int kb = kwb + t * 16;
        #pragma unroll
        for (int j = 0; j < 8; ++j)          // C-layout regs -> row-major LDS
            ps[(j + hi * 8) * 20 + ln] = sc[t][j];
        #pragma unroll
        for (int k4 = 0; k4 < 4; ++k4) {
            const v2f a = *(const v2f*)(ps + ln * 20 + k4 * 4 + 2 * hi);
            const int kr = kb + k4 * 4 + 2 * hi;              // even
            const float* vrow = Vp + hbD + (size_t)(kr >> 1) * (2 * DH_) + 2 * ln;
            #pragma unroll
            for (int n = 0; n < 4; ++n) {
                const v2f bb = *(const v2f*)(vrow + n * 32);  // (V[kr][d],V[kr+1][d])
                cacc[n] = __builtin_amdgcn_wmma_f32_16x16x4_f32(
                    false, a, false, bb, (short)0, cacc[n], false, false);
            }
        }
    }

    // ---- cross-wave ctx reduction ----
    #pragma unroll
    for (int n = 0; n < 4; ++n)
        #pragma unroll
        for (int j = 0; j < 8; ++j)
            red[wave][(j + hi * 8) * 68 + n * 16 + ln] = cacc[n][j];
    __syncthreads();

    {
        const int row  = tid >> 4;
        const int dseg = (tid & 15) * 4;
        float4 s = make_float4(0.f, 0.f, 0.f, 0.f);
        #pragma unroll
        for (int w = 0; w < 8; ++w) {
            const float4 v = *(const float4*)(&red[w][row * 68 + dseg]);
            s.x += v.x; s.y += v.y; s.z += v.z; s.w += v.w;
        }
        *(float4*)(ctx + ((size_t)(b * S_ + q0 + row)) * H_ + h * DH_ + dseg) = s;
    }
}

// ---------------------------------------------------------------------------
extern "C" void kernel_launch(void* const* d_in, const int* in_sizes, int n_in,
                              void* d_out, int out_size, void* d_ws, size_t ws_size,
                              hipStream_t stream)
{
    const float* X    = (const float*)d_in[0];
    const float* mask = (const float*)d_in[1];
    const float* Wq   = (const float*)d_in[2];
    const float* bq   = (const float*)d_in[3];
    const float* Wk   = (const float*)d_in[4];
    const float* bk   = (const float*)d_in[5];
    const float* Wv   = (const float*)d_in[6];
    const float* bv   = (const float*)d_in[7];

    const size_t perMat = (size_t)B_ * NH_ * S_ * DH_;   // 4M floats each
    float* Qh = (float*)d_ws;                            // 16 MB
    float* Kh = Qh + perMat;                             // 16 MB
    float* Vp = Kh + perMat;                             // 16 MB (pair-swizzled)
    float* Wp = Vp + perMat;                             // 12 MB (3 swizzled W)

    float* ctx  = (float*)d_out;                         // [B,S,H]
    float* attn = ctx + (size_t)B_ * S_ * H_;            // [B,NH,S,S]

    swizzle_w_kernel<<<dim3(3 * H_), 256, 0, stream>>>(Wq, Wk, Wv, Wp);

    qkv_proj_kernel<<<dim3((B_ * S_) / 16, 6), 256, 0, stream>>>(
        X, Wp, bq, bk, bv, Qh, Kh, Vp);

    attention_kernel<<<dim3(S_ / 16, NH_, B_), 256, 0, stream>>>(
        Qh, Kh, Vp, mask, ctx, attn);
}